// FeatureAdaption_27187142984231
// MI455X (gfx1250) — compile-verified
//
#include <hip/hip_runtime.h>
#include <hip/hip_bf16.h>
#include <math.h>

// Problem constants (match reference)
#define BB 2
#define CIN 256
#define COUT 256
#define HH 128
#define WW 128
#define HW (HH * WW)
#define KK 3
#define GG 4
#define OFFC (GG * 2 * KK * KK) // 72
#define NT 32                   // pixel tile per block

typedef __attribute__((ext_vector_type(2))) float v2f;
typedef __attribute__((ext_vector_type(4))) float v4f;
typedef __attribute__((ext_vector_type(8))) float v8f;

// ---------------------------------------------------------------------------
// Kernel 0: repack w_deform [COUT][CIN][3][3] -> wrep [tap][COUT][256] where
// within a row the k-order is [hi][kk][j]: pos = hi*128 + (ci>>2)*2 + (ci&1),
// hi = (ci>>1)&1. This lets a lane fetch A fragments for two consecutive
// k-steps with a single global_load_b128.
// ---------------------------------------------------------------------------
__global__ __launch_bounds__(256) void repack_w_kernel(
    const float* __restrict__ wd, float* __restrict__ wrep) {
  int idx = blockIdx.x * 256 + threadIdx.x;     // t*65536 + co*256 + ci
  if (idx >= 9 * COUT * CIN) return;
  int t = idx >> 16;                            // tap
  int r = idx & 65535;                          // co*256 + ci
  int co = r >> 8;
  int ci = r & 255;
  int pos = (((ci >> 1) & 1) << 7) + ((ci >> 2) << 1) + (ci & 1);
  wrep[(size_t)t * (COUT * CIN) + co * CIN + pos] = wd[(size_t)r * 9 + t];
}

// ---------------------------------------------------------------------------
// Kernel 1: 1x1 offset conv (coalesced over pixels, weight row wave-uniform)
// ---------------------------------------------------------------------------
__global__ __launch_bounds__(256) void offset_conv_kernel(
    const float* __restrict__ x, const float* __restrict__ w_off,
    const float* __restrict__ b_off, float* __restrict__ offs) {
  int blk = blockIdx.x;
  int chunk = blk & 63;             // 64 chunks of 256 pixels
  int bo = blk >> 6;                // 0 .. B*OFFC-1
  int b = bo / OFFC;
  int o = bo % OFFC;
  int p = chunk * 256 + threadIdx.x;
  const float* xb = x + (size_t)b * CIN * HW + p;
  const float* wrow = w_off + o * CIN;
  float acc = b_off[o];
#pragma unroll 8
  for (int c = 0; c < CIN; ++c)
    acc = fmaf(xb[(size_t)c * HW], wrow[c], acc);
  offs[((size_t)b * OFFC + o) * HW + p] = acc;
}

// ---------------------------------------------------------------------------
// Kernel 2: deformable conv GEMM. One block = 32 consecutive pixels (one b,h
// row segment). 8 waves; each wave owns cout tiles [m0,m0+16) and [m0+16,
// m0+32) x both 16-pixel N-tiles -> 4 accumulators. Per tap: VALU bilinear
// gather into a K-pair-packed, XOR-swizzled LDS B-tile, then 64 k-steps of
// f32 WMMA (processed two k-steps per b128 A-load).
// ---------------------------------------------------------------------------
__global__ __launch_bounds__(256) void deform_gemm_kernel(
    const float* __restrict__ x, const float* __restrict__ offs,
    const float* __restrict__ wrep, float* __restrict__ out) {
  // B tile: element (k, n) lives at float2 index (k/2)*NT + (n ^ (((k/2)&1)<<4)),
  // .x = even k, .y = odd k.  32 KB.
  __shared__ float bt[CIN * NT];

  const int tid = threadIdx.x;
  const int lane = tid & 31;
  const int lo = lane & 15;
  const int hi = lane >> 4;
  const int wave = tid >> 5;        // 0..7
  const int m0 = wave * 32;         // this wave covers cout [m0, m0+32)

  const int p0 = blockIdx.x * NT;   // flat pixel base (uniform b, h)
  const int b = p0 >> 14;           // / HW
  const int rem = p0 & 16383;
  const int h = rem >> 7;           // / WW
  const int w0 = rem & 127;

  // Sampling mapping: thread t -> pixel sn (0..31), channel block scb (0..7,
  // 32 channels each). A channel block stays inside one group (CG=64).
  const int sn = tid & 31;
  const int scb = tid >> 5;
  const int sg = scb >> 1;          // group 0..3
  const int wcol = w0 + sn;

  const float* xb = x + (size_t)b * CIN * HW;

  v8f acc00 = {};  // (m0,      n 0..15)
  v8f acc01 = {};  // (m0,      n 16..31)
  v8f acc10 = {};  // (m0+16,   n 0..15)
  v8f acc11 = {};  // (m0+16,   n 16..31)

  for (int tap = 0; tap < 9; ++tap) {
    const int ti = tap / 3;
    const int tj = tap % 3;

    // --- per-thread bilinear coordinates for (group sg, pixel sn) ---------
    const int och = (sg * 9 + tap) * 2;  // mmcv layout: (g, tap, dy/dx)
    const size_t obase = ((size_t)b * OFFC + och) * HW + (size_t)h * WW + wcol;
    const float dy = offs[obase];
    const float dx = offs[obase + HW];
    const float ys = (float)(h - 1 + ti) + dy;
    const float xs = (float)(wcol - 1 + tj) + dx;
    const float y0f = floorf(ys);
    const float x0f = floorf(xs);
    const float fy = ys - y0f;
    const float fx = xs - x0f;
    const int y0 = (int)y0f;
    const int x0 = (int)x0f;
    const float vy0 = (y0 >= 0 && y0 < HH) ? 1.f : 0.f;
    const float vy1 = (y0 + 1 >= 0 && y0 + 1 < HH) ? 1.f : 0.f;
    const float vx0 = (x0 >= 0 && x0 < WW) ? 1.f : 0.f;
    const float vx1 = (x0 + 1 >= 0 && x0 + 1 < WW) ? 1.f : 0.f;
    const float w00 = (1.f - fy) * (1.f - fx) * vy0 * vx0;
    const float w01 = (1.f - fy) * fx * vy0 * vx1;
    const float w10 = fy * (1.f - fx) * vy1 * vx0;
    const float w11 = fy * fx * vy1 * vx1;
    const int yc0 = min(max(y0, 0), HH - 1);
    const int yc1 = min(max(y0 + 1, 0), HH - 1);
    const int xc0 = min(max(x0, 0), WW - 1);
    const int xc1 = min(max(x0 + 1, 0), WW - 1);
    const int i00 = yc0 * WW + xc0;
    const int i01 = yc0 * WW + xc1;
    const int i10 = yc1 * WW + xc0;
    const int i11 = yc1 * WW + xc1;

    __syncthreads();                 // previous tap's GEMM done reading bt
#pragma unroll 4
    for (int ii = 0; ii < NT; ++ii) {
      const int c = scb * NT + ii;
      const float* xc = xb + (size_t)c * HW;
      const float v = w00 * xc[i00] + w01 * xc[i01] +
                      w10 * xc[i10] + w11 * xc[i11];
      const int kp = c >> 1;
      const int col = sn ^ ((kp & 1) << 4);    // bank swizzle
      bt[(kp * NT + col) * 2 + (c & 1)] = v;
    }
    __syncthreads();                 // B tile ready

    // --- GEMM: 4x D[16x16] += A[16x4] * B[4x16], K=256 (64 steps) ---------
    // A (repacked): lane (lo,hi) loads b128 at row(m)+hi*128+kk2*4 giving
    //   fragments {W[m][4k+2hi], W[m][4k+2hi+1]} for kk=2*kk2 and 2*kk2+1.
    // B: float2 at (2kk+hi)*NT + ((n0+lo) ^ (hi<<4)).
    const float* Wt = wrep + (size_t)tap * (COUT * CIN);
    const float* wa0 = Wt + (size_t)(m0 + lo) * CIN + hi * 128;
    const float* wa1 = wa0 + (size_t)16 * CIN;
    const float2* btp = (const float2*)bt;
    const int cs0 = lo ^ (hi << 4);          // swizzled col, n-tile 0
    const int cs1 = (16 + lo) ^ (hi << 4);   // swizzled col, n-tile 1
#pragma unroll 4
    for (int kk2 = 0; kk2 < 32; ++kk2) {
      const v4f a0q = *(const v4f*)(wa0 + kk2 * 4);
      const v4f a1q = *(const v4f*)(wa1 + kk2 * 4);
      const int kk = kk2 * 2;
      const v2f a0e = __builtin_shufflevector(a0q, a0q, 0, 1);
      const v2f a0o = __builtin_shufflevector(a0q, a0q, 2, 3);
      const v2f a1e = __builtin_shufflevector(a1q, a1q, 0, 1);
      const v2f a1o = __builtin_shufflevector(a1q, a1q, 2, 3);
      const float2 be0 = btp[(2 * kk + hi) * NT + cs0];
      const float2 be1 = btp[(2 * kk + hi) * NT + cs1];
      const float2 bo0 = btp[(2 * kk + 2 + hi) * NT + cs0];
      const float2 bo1 = btp[(2 * kk + 2 + hi) * NT + cs1];
      v2f bf;
      bf.x = be0.x; bf.y = be0.y;
      acc00 = __builtin_amdgcn_wmma_f32_16x16x4_f32(
          false, a0e, false, bf, (short)0, acc00, false, false);
      acc10 = __builtin_amdgcn_wmma_f32_16x16x4_f32(
          false, a1e, false, bf, (short)0, acc10, false, false);
      bf.x = be1.x; bf.y = be1.y;
      acc01 = __builtin_amdgcn_wmma_f32_16x16x4_f32(
          false, a0e, false, bf, (short)0, acc01, false, false);
      acc11 = __builtin_amdgcn_wmma_f32_16x16x4_f32(
          false, a1e, false, bf, (short)0, acc11, false, false);
      bf.x = bo0.x; bf.y = bo0.y;
      acc00 = __builtin_amdgcn_wmma_f32_16x16x4_f32(
          false, a0o, false, bf, (short)0, acc00, false, false);
      acc10 = __builtin_amdgcn_wmma_f32_16x16x4_f32(
          false, a1o, false, bf, (short)0, acc10, false, false);
      bf.x = bo1.x; bf.y = bo1.y;
      acc01 = __builtin_amdgcn_wmma_f32_16x16x4_f32(
          false, a0o, false, bf, (short)0, acc01, false, false);
      acc11 = __builtin_amdgcn_wmma_f32_16x16x4_f32(
          false, a1o, false, bf, (short)0, acc11, false, false);
    }
  }

  // --- ReLU + store. D layout: VGPR r -> M = r + 8*hi, N = lane%16 --------
  float* ob0 = out + (((size_t)b * COUT) * HH + h) * WW + w0 + lo;        // n0
  float* ob1 = ob0 + 16;                                                  // n1
#pragma unroll
  for (int r = 0; r < 8; ++r) {
    const int co0 = m0 + r + 8 * hi;
    const int co1 = m0 + 16 + r + 8 * hi;
    ob0[(size_t)co0 * HW] = fmaxf(acc00[r], 0.f);
    ob1[(size_t)co0 * HW] = fmaxf(acc01[r], 0.f);
    ob0[(size_t)co1 * HW] = fmaxf(acc10[r], 0.f);
    ob1[(size_t)co1 * HW] = fmaxf(acc11[r], 0.f);
  }
}

// ---------------------------------------------------------------------------
extern "C" void kernel_launch(void* const* d_in, const int* in_sizes, int n_in,
                              void* d_out, int out_size, void* d_ws,
                              size_t ws_size, hipStream_t stream) {
  (void)in_sizes; (void)n_in; (void)out_size; (void)ws_size;
  const float* x        = (const float*)d_in[0]; // [B,CIN,H,W]
  const float* w_offset = (const float*)d_in[1]; // [72,CIN]
  const float* b_offset = (const float*)d_in[2]; // [72]
  const float* w_deform = (const float*)d_in[3]; // [COUT,CIN,3,3]
  float* out = (float*)d_out;                    // [B,COUT,H,W]

  // Workspace layout: offsets [B,72,H,W] then repacked weights [9,COUT,CIN]
  float* offs = (float*)d_ws;                       // 2,359,296 floats
  float* wrep = offs + (size_t)BB * OFFC * HW;      //   589,824 floats

  repack_w_kernel<<<(9 * COUT * CIN + 255) / 256, 256, 0, stream>>>(w_deform,
                                                                    wrep);
  offset_conv_kernel<<<BB * OFFC * (HW / 256), 256, 0, stream>>>(
      x, w_offset, b_offset, offs);
  deform_gemm_kernel<<<(BB * HW) / NT, 256, 0, stream>>>(x, offs, wrep, out);
}